// ScaledDotProductAttention_11562051960991
// MI455X (gfx1250) — compile-verified
//
#include <hip/hip_runtime.h>

typedef __attribute__((ext_vector_type(2))) float v2f;
typedef __attribute__((ext_vector_type(8))) float v8f;

#define S_LEN 2048
#define D_DIM 64
#define QT 16            // query rows per workgroup
#define NWAVES 8
#define BLOCK (NWAVES * 32)
#define BH (4 * 8)
#define QTILES (S_LEN / QT)   // 128

__global__ __launch_bounds__(BLOCK) void sdpa_wmma_kernel(
    const float* __restrict__ q, const float* __restrict__ k,
    const float* __restrict__ v, const int* __restrict__ mask,
    float* __restrict__ out, float* __restrict__ attn)
{
    extern __shared__ float smem[];
    float* sc   = smem;                       // [QT][S_LEN] unnormalized exp scores
    float* oacc = smem + QT * S_LEN;          // [QT][D_DIM] output accumulator
    float* rinv = oacc + QT * D_DIM;          // [QT] 1/rowsum

    const int tid    = threadIdx.x;
    const int wave   = tid >> 5;
    const int lane   = tid & 31;
    const int lane16 = lane & 15;
    const int hi     = lane >> 4;             // 0 or 1 (half-wave select)

    const int bh = blockIdx.x >> 7;           // / QTILES
    const int qt = blockIdx.x & (QTILES - 1);
    const int b  = bh >> 3;
    const int q0 = qt * QT;

    const float* qbase = q + ((size_t)bh * S_LEN + q0) * D_DIM;
    const float* kbase = k + (size_t)bh * S_LEN * D_DIM;
    const float* vbase = v + (size_t)bh * S_LEN * D_DIM;
    const int*   mbase = mask + (size_t)b * S_LEN * S_LEN + (size_t)q0 * S_LEN;
    float* obase = out  + ((size_t)bh * S_LEN + q0) * D_DIM;
    float* abase = attn + (size_t)bh * S_LEN * S_LEN + (size_t)q0 * S_LEN;

    // ---- Q tile as 16 A-fragments (16x4 f32 layout: lanes0-15 K=0,1; lanes16-31 K=2,3)
    v2f aq[16];
#pragma unroll
    for (int c = 0; c < 16; ++c)
        aq[c] = *(const v2f*)(qbase + (size_t)lane16 * D_DIM + 4 * c + 2 * hi);

    // ---- scores = QK^T / 8, mask, into LDS
    for (int kt = wave; kt < QTILES; kt += NWAVES) {
        const int k0 = kt * QT;
        v8f acc = {};
#pragma unroll
        for (int c = 0; c < 16; ++c) {
            // B-fragment: B[d][n] = K[n][d]; same addressing pattern as A
            v2f bk = *(const v2f*)(kbase + (size_t)(k0 + lane16) * D_DIM + 4 * c + 2 * hi);
            acc = __builtin_amdgcn_wmma_f32_16x16x4_f32(
                false, aq[c], false, bk, (short)0, acc, false, false);
        }
        const int n = k0 + lane16;
#pragma unroll
        for (int r = 0; r < 8; ++r) {
            const int m = r + 8 * hi;                   // C/D layout: M = r + 8*hi, N = lane16
            float sv = acc[r] * 0.125f;                 // / sqrt(64)
            // mask is streamed once: non-temporal load, don't pollute L2 (K/V reuse lives there)
            if (__builtin_nontemporal_load(&mbase[(size_t)m * S_LEN + n]) == 0) sv = -1.0e9f;
            sc[m * S_LEN + n] = sv;
        }
    }
    __syncthreads();

    // ---- softmax stats: 16 threads per row, shuffle-reduce within 16-lane groups
    {
        const int row = tid >> 4;
        const int t   = tid & 15;
        float* srow = sc + row * S_LEN;
        float mx = -3.4e38f;
        for (int c = t; c < S_LEN; c += 16) mx = fmaxf(mx, srow[c]);
#pragma unroll
        for (int off = 1; off < 16; off <<= 1) mx = fmaxf(mx, __shfl_xor(mx, off, 32));
        float sum = 0.0f;
        for (int c = t; c < S_LEN; c += 16) {
            float e = __expf(srow[c] - mx);
            srow[c] = e;
            sum += e;
        }
#pragma unroll
        for (int off = 1; off < 16; off <<= 1) sum += __shfl_xor(sum, off, 32);
        if (t == 0) rinv[row] = 1.0f / sum;
    }
    for (int idx = tid; idx < QT * D_DIM; idx += BLOCK) oacc[idx] = 0.0f;
    __syncthreads();

    // ---- stream normalized probabilities to global: non-temporal (write-once traffic)
    for (int idx = tid; idx < QT * S_LEN; idx += BLOCK) {
        const int row = idx >> 11;                       // / S_LEN
        __builtin_nontemporal_store(sc[idx] * rinv[row], &abase[idx]);
    }

    // ---- out = P * V via WMMA (A from LDS scores, B from V), per-wave partials
    v8f ov[4];
#pragma unroll
    for (int g = 0; g < 4; ++g) ov[g] = (v8f){};

    for (int kt = wave; kt < QTILES; kt += NWAVES) {
        const int k0 = kt * QT;
#pragma unroll
        for (int c = 0; c < 4; ++c) {
            const int kk = k0 + 4 * c + 2 * hi;          // this half-wave's K rows
            v2f ap;
            ap.x = sc[lane16 * S_LEN + kk];
            ap.y = sc[lane16 * S_LEN + kk + 1];
#pragma unroll
            for (int g = 0; g < 4; ++g) {
                const int n = g * 16 + lane16;
                v2f bv;
                bv.x = vbase[(size_t)kk * D_DIM + n];
                bv.y = vbase[(size_t)(kk + 1) * D_DIM + n];
                ov[g] = __builtin_amdgcn_wmma_f32_16x16x4_f32(
                    false, ap, false, bv, (short)0, ov[g], false, false);
            }
        }
    }

    // ---- cross-wave reduction via LDS float atomics (ds_add_f32)
#pragma unroll
    for (int g = 0; g < 4; ++g)
#pragma unroll
        for (int r = 0; r < 8; ++r) {
            const int m = r + 8 * hi;
            const int n = g * 16 + lane16;
            atomicAdd(&oacc[m * D_DIM + n], ov[g][r]);
        }
    __syncthreads();

    for (int idx = tid; idx < QT * D_DIM; idx += BLOCK) {
        const int m = idx >> 6;                          // / D_DIM
        obase[idx] = oacc[idx] * rinv[m];
    }
}

extern "C" void kernel_launch(void* const* d_in, const int* in_sizes, int n_in,
                              void* d_out, int out_size, void* d_ws, size_t ws_size,
                              hipStream_t stream) {
    (void)in_sizes; (void)n_in; (void)out_size; (void)d_ws; (void)ws_size;
    const float* q    = (const float*)d_in[0];
    const float* k    = (const float*)d_in[1];
    const float* v    = (const float*)d_in[2];
    const int*   mask = (const int*)d_in[3];

    float* out  = (float*)d_out;                               // [B,H,S,D]
    float* attn = out + (size_t)BH * S_LEN * D_DIM;            // [B,H,S,S]

    const size_t smem = (size_t)(QT * S_LEN + QT * D_DIM + QT) * sizeof(float);
    (void)hipFuncSetAttribute((const void*)sdpa_wmma_kernel,
                              hipFuncAttributeMaxDynamicSharedMemorySize, (int)smem);

    dim3 grid(BH * QTILES);   // 4096 workgroups
    sdpa_wmma_kernel<<<grid, BLOCK, smem, stream>>>(q, k, v, mask, out, attn);
}